// EKVBlock_27144193310934
// MI455X (gfx1250) — compile-verified
//
#include <hip/hip_runtime.h>

// ---------------- constants (log2 domain, exp-sharing form) ----------------
#define EKV_ALPHA 0.0005625f
#define LOG2E     1.4426950408889634f
#define LN2       0.6931471805599453f
#define C1        36.99218053561445f      // LOG2E / (1.5*0.026): volts -> log2 units
#define VDC       3.699218053561445f      // VD * C1 (VD = 0.1)
#define ACLAMP2   46.98006953f            // 30*LOG2E + VDC : shared-exp input clamp
#define TCLAMP    1.06864746e13f          // e^30 = 2^(30*LOG2E) : t1 clamp
#define KVD       0.07698832f             // 2^-VDC = e^-(VD/nVT)

typedef __attribute__((ext_vector_type(2))) float    v2f;
typedef __attribute__((ext_vector_type(8))) float    v8f;
typedef __attribute__((ext_vector_type(4))) unsigned v4u;
typedef __attribute__((ext_vector_type(4))) int      v4i;
typedef __attribute__((ext_vector_type(8))) int      v8i;

#if __has_builtin(__builtin_amdgcn_tensor_load_to_lds) && \
    __has_builtin(__builtin_amdgcn_s_wait_tensorcnt)
#define USE_TDM 1
#endif

// load one prescaled/BN-affine input value with zero padding (affine BEFORE padding,
// matching reference: conv pads the already-transformed tensor with zeros).
__device__ __forceinline__ float load_vg(const float* __restrict__ src,
                                         const float* __restrict__ sc,
                                         const float* __restrict__ sh,
                                         int b, int c, int h, int w) {
    if ((unsigned)h < 32u && (unsigned)w < 32u) {
        float v = src[(((b << 6) + c) << 10) + (h << 5) + w];
        return fmaf(v, sc[c], sh[c]);
    }
    return 0.0f;
}

// one EKV term in log2 units with a SINGLE exp shared by both softplus branches:
//   e  = 2^min(a, 30*log2e + VDC)
//   l1 = log2(1 + min(e, e^30))          (== softplus(clip(a))/ln2)
//   l2 = log2(1 + e*2^-VDC)              (== softplus(clip(a - VD/nVT))/ln2)
// f = l1^2 - l2^2 ; exact w.r.t. the reference's independent +30 clips.
__device__ __forceinline__ float ekv_term(float a) {
    float e  = __builtin_amdgcn_exp2f(fminf(a, ACLAMP2));
    float l1 = __builtin_amdgcn_logf(fminf(e, TCLAMP) + 1.0f);
    float l2 = __builtin_amdgcn_logf(fmaf(e, KVD, 1.0f));
    return fmaf(l1, l1, -(l2 * l2));
}

// One EKV conv stage: each wave produces a 16-pixel x 16-channel f32 tile in the
// WMMA accumulator. Per (n, k0): lane computes 2 f-values in the ISA 16x4 f32
// A layout; B is a one-hot column selecting channel n, so v_wmma_f32_16x16x4_f32
// performs the exact K-reduction on the matrix pipe (co-executes with the VALU
// transcendental work, which is the roofline bottleneck here).
__global__ __launch_bounds__(256) void ekv_conv_kernel(
    const float* __restrict__ src,       // (4,64,32,32)
    const float* __restrict__ theta,     // (64,576)
    const float* __restrict__ in_scale,  // 64 (affine applied to src)
    const float* __restrict__ in_shift,  // 64
    const float* __restrict__ ls,        // scalar layer scale
    float* __restrict__ dst,             // (4,64,32,32) raw (pre-BN) output
    float* __restrict__ ch_sum,          // 64, fp32 atomics
    float* __restrict__ ch_sq)           // 64
{
    __shared__ float thp[16 * 576];      // raw theta tile (36 KB of 320 KB LDS)
    const int n0 = blockIdx.y << 4;

#ifdef USE_TDM
    if (threadIdx.x < 32) {              // wave 0 issues one TDM DMA for the theta tile
        unsigned long long ga = (unsigned long long)(uintptr_t)(theta + n0 * 576);
        unsigned lds_base = (unsigned)(uintptr_t)(void*)thp;  // generic LDS addr[31:0] = LDS offset
        // D# group0: count=1 | lds_addr | 57-bit global_addr | type=2
        v4u g0 = {1u, lds_base, (unsigned)ga,
                  (unsigned)((ga >> 32) & 0x01FFFFFFull) | 0x80000000u};
        // D# group1: data_size=4B; tensor 576x16 f32, stride 576; tile 576x16
        v8i g1 = {0x00020000,             // workgroup_mask=0 | data_size=2(4B)
                  0x02400000,             // tensor_dim0 low16 (576) at bits 63:48
                  0x00100000,             // tensor_dim1 low16 (16) at bits 95:80
                  0x02400000,             // tile_dim0 (576) at bits 127:112
                  16,                     // tile_dim1 = 16
                  576,                    // tensor_dim0_stride low32
                  0, 0};
        v4i g2 = {0, 0, 0, 0};
        v4i g3 = {0, 0, 0, 0};
#if __clang_major__ >= 23
        v8i g4 = {0, 0, 0, 0, 0, 0, 0, 0};
        __builtin_amdgcn_tensor_load_to_lds(g0, g1, g2, g3, g4, 0);
#else
        __builtin_amdgcn_tensor_load_to_lds(g0, g1, g2, g3, 0);
#endif
        __builtin_amdgcn_s_wait_tensorcnt(0);
    }
#else
    for (int i = threadIdx.x; i < 16 * 576; i += 256)
        thp[i] = theta[n0 * 576 + i];
#endif
    __syncthreads();

    const int lane = threadIdx.x & 31;
    const int wave = threadIdx.x >> 5;
    const int hi   = lane >> 4;          // 0: K = k0,k0+1   1: K = k0+2,k0+3
    const int ml   = lane & 15;          // A row / D column index
    const int pix  = ((blockIdx.x << 3) + wave) * 16 + ml;  // global pixel of A row
    const int b    = pix >> 10;
    const int h    = (pix >> 5) & 31;
    const int w    = pix & 31;

    v8f acc = {};                         // 16x16 f32 accumulator (8 VGPRs)

    for (int k0 = 0; k0 < 576; k0 += 4) {
        const int kA = k0 + 2 * hi;
        const int kB = kA + 1;
        // patch index k -> (c, r, s) with k = c*9 + r*3 + s
        const int c0 = kA / 9, rs0 = kA - c0 * 9, r0 = rs0 / 3, s0 = rs0 - r0 * 3;
        const int c1 = kB / 9, rs1 = kB - c1 * 9, r1 = rs1 / 3, s1 = rs1 - r1 * 3;

        const float vl0 = load_vg(src, in_scale, in_shift, b, c0, h + r0 - 1, w + s0 - 1) * C1;
        const float vl1 = load_vg(src, in_scale, in_shift, b, c1, h + r1 - 1, w + s1 - 1) * C1;

        // batch the 16 theta pairs first: ds_load_b64 x16 issue back-to-back,
        // latency covered by the transcendental chains below
        float2 tha[16];
        #pragma unroll
        for (int n = 0; n < 16; ++n)
            tha[n] = *(const float2*)&thp[n * 576 + kA];

        #pragma unroll
        for (int n = 0; n < 16; ++n) {
            float f0 = ekv_term(fmaf(tha[n].x, -C1, vl0));
            float f1 = ekv_term(fmaf(tha[n].y, -C1, vl1));
            const float bsel = (ml == n) ? 1.0f : 0.0f;  // one-hot column n
            v2f A = {f0, f1};
            v2f B = {bsel, bsel};
            acc = __builtin_amdgcn_wmma_f32_16x16x4_f32(
                false, A, false, B, (short)0, acc, false, false);
        }
    }

    // acc holds sums of (l1^2 - l2^2) in log2 units: scale by alpha*ln2^2*ls.
    const float osc = EKV_ALPHA * LN2 * LN2 * ls[0];
    const int ch    = n0 + ml;                          // D column = channel
    const int pbase = (((blockIdx.x << 3) + wave) << 4) + 8 * hi;  // D row base
    const int bb    = pbase >> 10;
    float* drow     = dst + ((((bb << 6) + ch) << 10) + (pbase & 1023));

    float s = 0.0f, q = 0.0f;
    #pragma unroll
    for (int r = 0; r < 8; ++r) {                       // VGPR r -> pixel pbase + r
        float v = acc[r] * osc;
        drow[r] = v;
        s += v;
        q = fmaf(v, v, q);
    }
    atomicAdd(&ch_sum[ch], s);
    atomicAdd(&ch_sq[ch], q);
}

// zero BN accumulators; broadcast prescale scalars into per-channel affine arrays
__global__ void prep0(const float* __restrict__ ps, const float* __restrict__ pb,
                      float* __restrict__ st) {
    int t = threadIdx.x;
    st[t] = 0.0f;                         // [0..255] = sum1,sq1,sum2,sq2
    if (t < 64) { st[256 + t] = ps[0]; st[320 + t] = pb[0]; }
}

__global__ void bnprep(const float* __restrict__ sum, const float* __restrict__ sq,
                       const float* __restrict__ gamma, const float* __restrict__ beta,
                       float* __restrict__ scale, float* __restrict__ shift) {
    int c = threadIdx.x;
    if (c < 64) {
        float mean = sum[c] * (1.0f / 4096.0f);
        float var  = fmaf(-mean, mean, sq[c] * (1.0f / 4096.0f));
        float sc   = gamma[c] * rsqrtf(var + 1e-5f);
        scale[c] = sc;
        shift[c] = fmaf(-mean, sc, beta[c]);
    }
}

__global__ void finalize(const float* __restrict__ y2, const float* __restrict__ scale,
                         const float* __restrict__ shift, const float* __restrict__ x,
                         float* __restrict__ out) {
    int i = blockIdx.x * 256 + threadIdx.x;
    if (i < 4 * 64 * 32 * 32) {
        int ch = (i >> 10) & 63;
        out[i] = fmaf(y2[i], scale[ch], shift[ch]) + x[i];
    }
}

extern "C" void kernel_launch(void* const* d_in, const int* in_sizes, int n_in,
                              void* d_out, int out_size, void* d_ws, size_t ws_size,
                              hipStream_t stream) {
    const float* x      = (const float*)d_in[0];
    const float* theta1 = (const float*)d_in[1];
    const float* theta2 = (const float*)d_in[2];
    const float* ps     = (const float*)d_in[3];
    const float* pb     = (const float*)d_in[4];
    const float* ls1    = (const float*)d_in[5];
    const float* ls2    = (const float*)d_in[6];
    const float* g1     = (const float*)d_in[7];
    const float* b1     = (const float*)d_in[8];
    const float* g2     = (const float*)d_in[9];
    const float* b2     = (const float*)d_in[10];

    float* ws = (float*)d_ws;
    float* y1 = ws;                       // 262144
    float* y2 = ws + 262144;              // 262144
    float* st = ws + 524288;              // 640 stats/affine floats
    float *sum1 = st,       *sq1 = st + 64,  *sum2 = st + 128, *sq2 = st + 192;
    float *sc0  = st + 256, *sh0 = st + 320;
    float *sc1  = st + 384, *sh1 = st + 448;
    float *sc2  = st + 512, *sh2 = st + 576;
    float* out  = (float*)d_out;

    dim3 grid(32, 4);                     // 32 m-groups (8 waves x 16 pixels) x 4 n-tiles

    prep0<<<1, 256, 0, stream>>>(ps, pb, st);
    ekv_conv_kernel<<<grid, 256, 0, stream>>>(x,  theta1, sc0, sh0, ls1, y1, sum1, sq1);
    bnprep<<<1, 64, 0, stream>>>(sum1, sq1, g1, b1, sc1, sh1);
    ekv_conv_kernel<<<grid, 256, 0, stream>>>(y1, theta2, sc1, sh1, ls2, y2, sum2, sq2);
    bnprep<<<1, 64, 0, stream>>>(sum2, sq2, g2, b2, sc2, sh2);
    finalize<<<1024, 256, 0, stream>>>(y2, sc2, sh2, x, out);
}